// IHCLowpassFilter_28613072126391
// MI455X (gfx1250) — compile-verified
//
#include <hip/hip_runtime.h>
#include <hip/hip_bf16.h>

// ---------------------------------------------------------------------------
// IHC lowpass: 7 cascaded 1st-order IIRs over 800 rows x 80000 samples, then
// decimate by 2. One workgroup per row; entire row resident in 320KB LDS.
// CDNA5 paths: async global->LDS loads (ASYNCcnt), async LDS->global stores,
// wave32 shuffle scans for the affine-recurrence carries.
// ---------------------------------------------------------------------------

#define N_SAMP   80000
#define CHUNK    100
#define NTHREADS 800
#define NWAVES   (NTHREADS / 32)
#define N_OUT    40000
#define OUT_PER_T (N_OUT / NTHREADS)   // 50

// Filter coefficients (bilinear transform), computed exactly as the reference.
constexpr double PI_D   = 3.14159265358979323846;
constexpr double W_D    = 2.0 * PI_D * 4800.0;
constexpr double CC_D   = 2.0 * 40000.0;
constexpr float  C1F    = (float)((CC_D - W_D) / (CC_D + W_D)); // feedback
constexpr float  C2F    = (float)(W_D / (CC_D + W_D));          // feedforward

constexpr float cpow(float c, int n) {
    float r = 1.0f;
    for (int i = 0; i < n; ++i) r *= c;
    return r;
}
constexpr float A_CHUNK = cpow(C1F, CHUNK);  // C1^100 (~3.6e-35, still normal)

// ---- CDNA5 async-to/from-LDS feature detection -----------------------------
#if defined(__has_builtin)
#  if __has_builtin(__builtin_amdgcn_global_load_async_to_lds_b128)
#    define HAVE_ASYNC_LOAD 1
#  endif
#  if __has_builtin(__builtin_amdgcn_global_store_async_from_lds_b32)
#    define HAVE_ASYNC_STORE 1
#  endif
#  if __has_builtin(__builtin_amdgcn_s_wait_asynccnt)
#    define HAVE_WAIT_ASYNC_BUILTIN 1
#  endif
#endif

#if defined(HAVE_ASYNC_LOAD) || defined(HAVE_ASYNC_STORE)
typedef int v4i __attribute__((vector_size(16)));
typedef __attribute__((address_space(1))) v4i  as1_v4i;   // global 16B granule
typedef __attribute__((address_space(1))) int  as1_int;   // global 4B granule
typedef __attribute__((address_space(3))) v4i  as3_v4i;   // LDS 16B granule
typedef __attribute__((address_space(3))) int  as3_int;   // LDS 4B granule
#endif

__device__ __forceinline__ void wait_async_zero() {
#if defined(HAVE_WAIT_ASYNC_BUILTIN)
    __builtin_amdgcn_s_wait_asynccnt(0);
#elif defined(HAVE_ASYNC_LOAD) || defined(HAVE_ASYNC_STORE)
    asm volatile("s_wait_asynccnt 0" ::: "memory");
#endif
}

// Block-level exclusive scan of affine elements (a, b) : y -> a*y + b, where
// a == A_CHUNK for every element. Returns yin = exclusive-prefix b, i.e. the
// recurrence state entering this thread's chunk. Uses wave32 shuffles +
// per-wave LDS aggregates. Contains 3 __syncthreads (uniform for all threads).
__device__ __forceinline__ float block_scan_carry(float carry, int lane, int wid,
                                                  float* wA, float* wB) {
    float a = A_CHUNK, b = carry;
#pragma unroll
    for (int d = 1; d < 32; d <<= 1) {
        float au = __shfl_up(a, (unsigned)d, 32);
        float bu = __shfl_up(b, (unsigned)d, 32);
        if (lane >= d) { b = fmaf(a, bu, b); a = a * au; }
    }
    const float a_inc = a, b_inc = b;

    __syncthreads();                      // protect scratch reuse across stages
    if (lane == 31) { wA[wid] = a_inc; wB[wid] = b_inc; }
    __syncthreads();

    if (wid == 0) {
        float wa = (lane < NWAVES) ? wA[lane] : 1.0f;
        float wb = (lane < NWAVES) ? wB[lane] : 0.0f;
#pragma unroll
        for (int d = 1; d < 32; d <<= 1) {
            float au = __shfl_up(wa, (unsigned)d, 32);
            float bu = __shfl_up(wb, (unsigned)d, 32);
            if (lane >= d) { wb = fmaf(wa, bu, wb); wa = wa * au; }
        }
        if (lane < NWAVES) { wA[lane] = wa; wB[lane] = wb; }
    }
    __syncthreads();

    float bW = 0.0f;
    if (wid > 0) { bW = wB[wid - 1]; }
    // lane-exclusive within wave
    float aLx = __shfl_up(a_inc, 1u, 32);
    float bLx = __shfl_up(b_inc, 1u, 32);
    if (lane == 0) { aLx = 1.0f; bLx = 0.0f; }
    // total exclusive prefix applied to state 0: yin = aLx*bW + bLx
    return fmaf(aLx, bW, bLx);
}

__global__ __launch_bounds__(NTHREADS)
void IHCLowpassFilter_28613072126391_kernel(const float* __restrict__ x,
                                            float* __restrict__ out) {
    extern __shared__ float smem[];
    float* data = smem;                 // 80000 floats: the whole row
    float* wA   = smem + N_SAMP;        // 32 floats: wave aggregate 'a'
    float* wB   = smem + N_SAMP + 32;   // 32 floats: wave aggregate 'b'

    const int t     = threadIdx.x;
    const int lane  = t & 31;
    const int wid   = t >> 5;
    const int start = t * CHUNK;
    const float c1 = C1F, c2 = C2F;

    const float* __restrict__ xrow = x + (size_t)blockIdx.x * N_SAMP;

    // ---- Load entire row into LDS (async global->LDS, 16B granules) -------
#if defined(HAVE_ASYNC_LOAD)
#pragma unroll
    for (int j = 0; j < CHUNK / 4; ++j) {
        __builtin_amdgcn_global_load_async_to_lds_b128(
            (as1_v4i*)(xrow + start + 4 * j),
            (as3_v4i*)(data + start + 4 * j),
            0, 0);
    }
    wait_async_zero();                  // own chunk resident
#else
#pragma unroll
    for (int j = 0; j < CHUNK / 4; ++j) {
        const float4 v = ((const float4*)(xrow + start))[j];
        ((float4*)(data + start))[j] = v;
    }
#endif
    __syncthreads();                    // all chunks visible

    // ---- Stage 1: u = C2*(x[t]+x[t-1]) needs left neighbor's raw sample ---
    float xm1 = (t > 0) ? data[start - 1] : 0.0f;
    __syncthreads();                    // boundary reads before in-place write

    float y = 0.0f;
#pragma unroll 4
    for (int i = 0; i < CHUNK; ++i) {
        const float v = data[start + i];
        const float u = c2 * (v + xm1);
        y = fmaf(c1, y, u);
        data[start + i] = y;            // partial solution (zero entering state)
        xm1 = v;
    }
    float yin = block_scan_carry(y, lane, wid, wA, wB);

    // ---- Stages 2..7: correct previous stage on the fly, run next stage ---
#pragma unroll 1
    for (int s = 0; s < 6; ++s) {
        float vm1 = yin;                // corrected y_{s}[start-1] == yin
        float p   = c1;                 // C1^(i - start + 1)
        float yy  = 0.0f;
#pragma unroll 4
        for (int i = 0; i < CHUNK; ++i) {
            const float v = fmaf(p, yin, data[start + i]); // corrected prev stage
            p *= c1;
            const float u = c2 * (v + vm1);
            yy = fmaf(c1, yy, u);
            data[start + i] = yy;       // partial solution of this stage
            vm1 = v;
        }
        yin = block_scan_carry(yy, lane, wid, wA, wB);
    }

    // ---- Final correction of stage-7 partials, in place --------------------
    {
        float p = c1;
#pragma unroll 4
        for (int i = 0; i < CHUNK; ++i) {
            data[start + i] = fmaf(p, yin, data[start + i]);
            p *= c1;
        }
    }
    __syncthreads();

    // ---- Decimate by 2, coalesced across threads (async LDS->global) ------
    float* __restrict__ orow = out + (size_t)blockIdx.x * N_OUT;
#pragma unroll
    for (int j = 0; j < OUT_PER_T; ++j) {
        const int o = t + NTHREADS * j;
#if defined(HAVE_ASYNC_STORE)
        __builtin_amdgcn_global_store_async_from_lds_b32(
            (as1_int*)(orow + o),
            (as3_int*)(data + 2 * o),
            0, 0);
#else
        orow[o] = data[2 * o];
#endif
    }
#if defined(HAVE_ASYNC_STORE)
    wait_async_zero();                  // (S_ENDPGM would also drain counters)
#endif
}

extern "C" void kernel_launch(void* const* d_in, const int* in_sizes, int n_in,
                              void* d_out, int out_size, void* d_ws, size_t ws_size,
                              hipStream_t stream) {
    (void)n_in; (void)out_size; (void)d_ws; (void)ws_size;
    const float* x  = (const float*)d_in[0];
    float* out      = (float*)d_out;
    const int rows  = in_sizes[0] / N_SAMP;          // 16*50 = 800

    const size_t shmem = (size_t)(N_SAMP + 64) * sizeof(float);  // 320,256 B
    (void)hipFuncSetAttribute((const void*)IHCLowpassFilter_28613072126391_kernel,
                              hipFuncAttributeMaxDynamicSharedMemorySize, (int)shmem);

    IHCLowpassFilter_28613072126391_kernel<<<rows, NTHREADS, shmem, stream>>>(x, out);
}